// BasicBlock_88364657148229
// MI455X (gfx1250) — compile-verified
//
#include <hip/hip_runtime.h>

typedef __attribute__((ext_vector_type(2))) float v2f;
typedef __attribute__((ext_vector_type(8))) float v8f;

#define NBLK_STATS 1024

__device__ __forceinline__ v8f wmma_f32(v2f a, v2f b, v8f c) {
  // V_WMMA_F32_16X16X4_F32: D(16x16) = A(16x4) * B(4x16) + C
  return __builtin_amdgcn_wmma_f32_16x16x4_f32(false, a, false, b, (short)0, c,
                                               false, false);
}

// ---------------------------------------------------------------- utilities
__global__ __launch_bounds__(256) void zero_kernel(float* __restrict__ p, long long n) {
  long long i = (long long)blockIdx.x * blockDim.x + threadIdx.x;
  long long st = (long long)gridDim.x * blockDim.x;
  for (; i < n; i += st) p[i] = 0.0f;
}

__global__ __launch_bounds__(256) void count_kernel(const int* __restrict__ e,
                                                    int* __restrict__ cnt, int total) {
  int i = blockIdx.x * blockDim.x + threadIdx.x;
  int st = gridDim.x * blockDim.x;
  for (; i < total; i += st) atomicAdd(&cnt[e[i]], 1);
}

// -------------------------------------------------- layer 1 GEMM (K=64) via WMMA
// x: [B, 8, N] (channel-major). features k = vslot*8 + c.  W1: [128, 64].
// Output scatter-add into acc [B, N, 16] (o = vslot2*16 + co -> vertex elems[m, vslot2], ch co).
__global__ __launch_bounds__(256) void gemm1_kernel(
    const float* __restrict__ x, const int* __restrict__ elems,
    const float* __restrict__ W1, const float* __restrict__ b1,
    float* __restrict__ acc, int N, int M, int mblkPerB) {
  __shared__ float Wt[64][128];     // Wt[k][o] = W1[o*64 + k]   (32 KB)
  __shared__ int ids_t[8][8][16];   // [wave][vslot][row]         (4 KB)
  const int tid = threadIdx.x;
  for (int i = tid; i < 64 * 128; i += 256) {
    int k = i >> 7, o = i & 127;
    Wt[k][o] = W1[o * 64 + k];      // conflict-free LDS writes (consecutive o)
  }
  const int wave = tid >> 5, lane = tid & 31;
  const int b = blockIdx.x / mblkPerB;
  const int m0 = __builtin_amdgcn_readfirstlane(
      (blockIdx.x % mblkPerB) * 128 + wave * 16);       // wave-uniform -> SGPR
  const int r = lane & 15;          // my GEMM row within tile
  const int hi = lane >> 4;         // lane half (A holds k pairs {0,1} vs {2,3} mod 4)
  const int hi2 = hi << 1;
  const int mrow = m0 + r;
  const int mcl = mrow < M ? mrow : (M - 1);
  int ids[8];
  const int* ep = elems + (long long)mcl * 8;
#pragma unroll
  for (int t = 0; t < 8; ++t) ids[t] = ep[t];
  if (hi == 0) {
#pragma unroll
    for (int t = 0; t < 8; ++t) ids_t[wave][t][r] = ids[t];
  }
  __syncthreads();

  // Gather A fragments: 16 k-steps, 2 floats/lane each.
  const float* xb = x + (long long)b * 8 * N;
  v2f A[16];
#pragma unroll
  for (int s = 0; s < 16; ++s) {
    const int vs = s >> 1;                    // compile-time vertex slot
    const unsigned c0 = ((s & 1) << 2) + hi2; // channel of k0 = 4s + hi2
    const unsigned vid = (unsigned)ids[vs];
    A[s].x = xb[c0 * (unsigned)N + vid];
    A[s].y = xb[(c0 + 1) * (unsigned)N + vid];
  }

  float* accb = acc + (long long)b * N * 16;
  const unsigned col = lane & 15;
  const bool fullTile = (m0 + 16 <= M);       // scalar branch (m0 is SGPR)
#pragma unroll
  for (int j = 0; j < 8; ++j) {               // 8 output tiles of 16 (o = 16j + col)
    v8f d;
    const float bias = b1[j * 16 + col];
#pragma unroll
    for (int i = 0; i < 8; ++i) d[i] = bias;  // bias uniform over tile rows
#pragma unroll
    for (int s = 0; s < 16; ++s) {
      v2f bf;
      bf.x = Wt[4 * s + hi2][j * 16 + col];
      bf.y = Wt[4 * s + hi2 + 1][j * 16 + col];
      d = wmma_f32(A[s], bf, d);
    }
    // D layout: VGPR i -> row (8*hi + i), col = lane%16. Scatter to vertex elems[m, j].
    const int4* ip = (const int4*)&ids_t[wave][j][hi * 8];
    const int4 q0 = ip[0];
    const int4 q1 = ip[1];
    const int rid[8] = {q0.x, q0.y, q0.z, q0.w, q1.x, q1.y, q1.z, q1.w};
    if (fullTile) {
#pragma unroll
      for (int i = 0; i < 8; ++i) {
        const unsigned off = (unsigned)rid[i] * 16u + col;
        (void)__hip_atomic_fetch_add(accb + off, d[i], __ATOMIC_RELAXED,
                                     __HIP_MEMORY_SCOPE_AGENT);
      }
    } else {
#pragma unroll
      for (int i = 0; i < 8; ++i) {
        if (m0 + hi * 8 + i < M) {
          const unsigned off = (unsigned)rid[i] * 16u + col;
          (void)__hip_atomic_fetch_add(accb + off, d[i], __ATOMIC_RELAXED,
                                       __HIP_MEMORY_SCOPE_AGENT);
        }
      }
    }
  }
}

// -------------------------------------------------- layer 2 GEMM (K=128) via WMMA
// y: [B, N, 16] (vertex-major). features k = vslot*16 + c.  W2: [128, 128].
__global__ __launch_bounds__(256) void gemm2_kernel(
    const float* __restrict__ y, const int* __restrict__ elems,
    const float* __restrict__ W2, const float* __restrict__ b2,
    float* __restrict__ acc, int N, int M, int mblkPerB) {
  __shared__ float Wt[128][128];    // 64 KB
  __shared__ int ids_t[8][8][16];
  const int tid = threadIdx.x;
  for (int i = tid; i < 128 * 128; i += 256) {
    int k = i >> 7, o = i & 127;
    Wt[k][o] = W2[o * 128 + k];
  }
  const int wave = tid >> 5, lane = tid & 31;
  const int b = blockIdx.x / mblkPerB;
  const int m0 = __builtin_amdgcn_readfirstlane(
      (blockIdx.x % mblkPerB) * 128 + wave * 16);
  const int r = lane & 15;
  const int hi = lane >> 4;
  const int hi2 = hi << 1;
  const int mrow = m0 + r;
  const int mcl = mrow < M ? mrow : (M - 1);
  int ids[8];
  const int* ep = elems + (long long)mcl * 8;
#pragma unroll
  for (int t = 0; t < 8; ++t) ids[t] = ep[t];
  if (hi == 0) {
#pragma unroll
    for (int t = 0; t < 8; ++t) ids_t[wave][t][r] = ids[t];
  }
  __syncthreads();

  const float* yb = y + (long long)b * N * 16;
  v2f A[32];
#pragma unroll
  for (int s = 0; s < 32; ++s) {
    const int vs = s >> 2;                    // compile-time vertex slot
    const unsigned c0 = ((s & 3) << 2) + hi2; // channel 0..15
    const unsigned base = (unsigned)ids[vs] * 16u;
    A[s].x = yb[base + c0];
    A[s].y = yb[base + c0 + 1];
  }

  float* accb = acc + (long long)b * N * 16;
  const unsigned col = lane & 15;
  const bool fullTile = (m0 + 16 <= M);
#pragma unroll
  for (int j = 0; j < 8; ++j) {
    v8f d;
    const float bias = b2[j * 16 + col];
#pragma unroll
    for (int i = 0; i < 8; ++i) d[i] = bias;
#pragma unroll
    for (int s = 0; s < 32; ++s) {
      v2f bf;
      bf.x = Wt[4 * s + hi2][j * 16 + col];
      bf.y = Wt[4 * s + hi2 + 1][j * 16 + col];
      d = wmma_f32(A[s], bf, d);
    }
    const int4* ip = (const int4*)&ids_t[wave][j][hi * 8];
    const int4 q0 = ip[0];
    const int4 q1 = ip[1];
    const int rid[8] = {q0.x, q0.y, q0.z, q0.w, q1.x, q1.y, q1.z, q1.w};
    if (fullTile) {
#pragma unroll
      for (int i = 0; i < 8; ++i) {
        const unsigned off = (unsigned)rid[i] * 16u + col;
        (void)__hip_atomic_fetch_add(accb + off, d[i], __ATOMIC_RELAXED,
                                     __HIP_MEMORY_SCOPE_AGENT);
      }
    } else {
#pragma unroll
      for (int i = 0; i < 8; ++i) {
        if (m0 + hi * 8 + i < M) {
          const unsigned off = (unsigned)rid[i] * 16u + col;
          (void)__hip_atomic_fetch_add(accb + off, d[i], __ATOMIC_RELAXED,
                                       __HIP_MEMORY_SCOPE_AGENT);
        }
      }
    }
  }
}

// ---------------------------- scatter-mean divide + per-block BN partial sums
// y layout [B, N, 16]; thread's channel = tid % 16 (grid/block multiples of 16).
__global__ __launch_bounds__(256) void stats_kernel(float* __restrict__ y,
                                                    const int* __restrict__ cnt,
                                                    float* __restrict__ partial,
                                                    int N, long long total) {
  __shared__ float s_sum[256], s_sq[256];
  const int tid = threadIdx.x;
  float sum = 0.0f, sq = 0.0f;
  const long long st = (long long)gridDim.x * blockDim.x;  // multiple of 16
  for (long long i = (long long)blockIdx.x * 256 + tid; i < total; i += st) {
    const int n = (int)((i >> 4) % N);
    const float c = (float)cnt[n];
    const float v = y[i] / fmaxf(c, 1.0f);
    y[i] = v;
    sum += v;
    sq += v * v;
  }
  s_sum[tid] = sum;
  s_sq[tid] = sq;
  __syncthreads();
  for (int off = 128; off >= 16; off >>= 1) {
    if (tid < off) {
      s_sum[tid] += s_sum[tid + off];
      s_sq[tid] += s_sq[tid + off];
    }
    __syncthreads();
  }
  if (tid < 16) {
    partial[blockIdx.x * 32 + tid] = s_sum[tid];
    partial[blockIdx.x * 32 + 16 + tid] = s_sq[tid];
  }
}

// ------------------------------- combine partials -> fused scale/shift (16 ch)
__global__ void bn_finalize(const float* __restrict__ partial, int nblk,
                            const float* __restrict__ gamma,
                            const float* __restrict__ beta, float cntBN,
                            float* __restrict__ ss) {
  const int c = threadIdx.x;
  if (c < 16) {
    float sum = 0.0f, sq = 0.0f;
    for (int bl = 0; bl < nblk; ++bl) {
      sum += partial[bl * 32 + c];
      sq += partial[bl * 32 + 16 + c];
    }
    const float mean = sum / cntBN;
    const float var = sq / cntBN - mean * mean;
    const float inv = rsqrtf(var + 1e-5f);
    const float sc = gamma[c] * inv;
    ss[c] = sc;
    ss[16 + c] = beta[c] - mean * sc;
  }
}

__global__ __launch_bounds__(256) void bn_apply_relu(float* __restrict__ y,
                                                     const float* __restrict__ ss,
                                                     long long total) {
  long long i = (long long)blockIdx.x * blockDim.x + threadIdx.x;
  const long long st = (long long)gridDim.x * blockDim.x;
  for (; i < total; i += st) {
    const int ch = (int)(i & 15);
    const float v = y[i] * ss[ch] + ss[16 + ch];
    y[i] = fmaxf(v, 0.0f);
  }
}

// ------------------------ BN2 apply + skip GEMV + ReLU + transpose to [B,Co,N]
__global__ __launch_bounds__(256) void final_kernel(
    const float* __restrict__ x, const float* __restrict__ y2,
    const float* __restrict__ ss2, const float* __restrict__ Wskip,
    const float* __restrict__ bskip, float* __restrict__ out, int N, int BN) {
  int idx = blockIdx.x * blockDim.x + threadIdx.x;
  const int st = gridDim.x * blockDim.x;
  for (; idx < BN; idx += st) {
    const int b = idx / N, n = idx % N;
    const float* xb = x + (long long)b * 8 * N + n;
    float xv[8];
#pragma unroll
    for (int c = 0; c < 8; ++c) xv[c] = xb[(long long)c * N];
    const float* yv = y2 + (long long)idx * 16;
    float* ob = out + (long long)b * 16 * N + n;
#pragma unroll
    for (int co = 0; co < 16; ++co) {
      float skip = bskip[co];
#pragma unroll
      for (int c = 0; c < 8; ++c) skip += Wskip[co * 8 + c] * xv[c];
      const float v = yv[co] * ss2[co] + ss2[16 + co] + skip;
      ob[(long long)co * N] = fmaxf(v, 0.0f);
    }
  }
}

// ----------------------------------------------------------------- launcher
extern "C" void kernel_launch(void* const* d_in, const int* in_sizes, int n_in,
                              void* d_out, int out_size, void* d_ws, size_t ws_size,
                              hipStream_t stream) {
  (void)n_in; (void)out_size; (void)ws_size;
  const float* x      = (const float*)d_in[0];
  const int*   elems  = (const int*)d_in[1];
  const float* W1     = (const float*)d_in[2];
  const float* b1     = (const float*)d_in[3];
  const float* gamma1 = (const float*)d_in[4];
  const float* beta1  = (const float*)d_in[5];
  const float* W2     = (const float*)d_in[6];
  const float* b2     = (const float*)d_in[7];
  const float* gamma2 = (const float*)d_in[8];
  const float* beta2  = (const float*)d_in[9];
  const float* Wskip  = (const float*)d_in[10];
  const float* bskip  = (const float*)d_in[11];
  float* out = (float*)d_out;

  const int Bb = 4, Cin = 8, Co = 16;         // reference setup shapes
  const int M = in_sizes[1] / 8;
  const int N = in_sizes[0] / (Cin * Bb);
  const long long L1 = (long long)Bb * N * Co;

  float* ws   = (float*)d_ws;
  float* acc1 = ws;                            // [B,N,16] -> becomes y1 (post BN+relu)
  float* acc2 = ws + L1;                       // [B,N,16] -> becomes y2 (post mean-div)
  int*   cnt  = (int*)(ws + 2 * L1);           // [N]
  float* part = ws + 2 * L1 + N;               // [NBLK_STATS * 32]
  float* ss1  = part + (long long)NBLK_STATS * 32;  // scale[16], shift[16]
  float* ss2  = ss1 + 32;

  const long long zeroWords = 2 * L1 + N;
  zero_kernel<<<4096, 256, 0, stream>>>(ws, zeroWords);

  const int etotal = M * 8;
  count_kernel<<<(etotal + 255) / 256, 256, 0, stream>>>(elems, cnt, etotal);

  const int mblkPerB = (M + 127) / 128;
  const long long total16 = L1;
  const float cntBN = (float)((long long)Bb * N);

  gemm1_kernel<<<Bb * mblkPerB, 256, 0, stream>>>(x, elems, W1, b1, acc1, N, M, mblkPerB);
  stats_kernel<<<NBLK_STATS, 256, 0, stream>>>(acc1, cnt, part, N, total16);
  bn_finalize<<<1, 64, 0, stream>>>(part, NBLK_STATS, gamma1, beta1, cntBN, ss1);
  bn_apply_relu<<<4096, 256, 0, stream>>>(acc1, ss1, total16);

  gemm2_kernel<<<Bb * mblkPerB, 256, 0, stream>>>(acc1, elems, W2, b2, acc2, N, M, mblkPerB);
  stats_kernel<<<NBLK_STATS, 256, 0, stream>>>(acc2, cnt, part, N, total16);
  bn_finalize<<<1, 64, 0, stream>>>(part, NBLK_STATS, gamma2, beta2, cntBN, ss2);

  const int BN = Bb * N;
  final_kernel<<<(BN + 255) / 256, 256, 0, stream>>>(x, acc2, ss2, Wskip, bskip, out, N, BN);
}